// MotifPool_84318797955332
// MI455X (gfx1250) — compile-verified
//
#include <hip/hip_runtime.h>
#include <hip/hip_bf16.h>

#define NUM_GRAPHS 8192
#define HIDDEN     256
#define HEADS      8
#define HEAD_DIM   32
#define N_ATOMS    400000
#define N_CLIQUES  150000
#define N_A2C      800000

typedef __attribute__((ext_vector_type(2))) float v2f;
typedef __attribute__((ext_vector_type(8))) float v8f;

// ---------------- K0: zero a float buffer ----------------
__global__ void k0_zero(float* __restrict__ p, long n) {
    long i = (long)blockIdx.x * blockDim.x + threadIdx.x;
    long stride = (long)gridDim.x * blockDim.x;
    for (; i < n; i += stride) p[i] = 0.0f;
}

// ---------------- K1: scatter-add atoms -> cliques ----------------
// One block per edge; 256 threads = 256 channels. ssum is L2-resident (150MB < 192MB L2),
// so the f32 atomics resolve in-cache.
__global__ __launch_bounds__(256) void k1_scatter(
    const float* __restrict__ x, const int* __restrict__ a2c,
    float* ssum, float* cnt)
{
    int e = blockIdx.x;
    int row = a2c[e];               // atom index
    int col = a2c[N_A2C + e];       // clique index
    int t = threadIdx.x;
    float v = x[(long)row * HIDDEN + t];
    atomicAdd(&ssum[(long)col * HIDDEN + t], v);
    if (t == 0) atomicAdd(&cnt[col], 1.0f);
}

// ---------------- K2: fused scatter-mean-div + GEMM(WMMA f32) + bias + relu
// ---------------- + residual + per-head MLP score --------------------------
// Block = 256 threads (8 waves), owns 16 clique rows. Each wave computes two
// 16x16 output tiles with V_WMMA_F32_16X16X4_F32 over K=256.
__global__ __launch_bounds__(256) void k2_gemm_score(
    const float* __restrict__ xc_in,    // original x_clique [Nc,256]
    const float* __restrict__ W_lin,    // [256,256] row-major W[k][n]
    const float* __restrict__ b_lin,    // [256]
    const float* __restrict__ W1g,      // [8,32,64]
    const float* __restrict__ b1g,      // [8,64]
    const float* __restrict__ W2g,      // [8,64]
    const float* __restrict__ b2g,      // [8]
    const float* ssum,                  // [Nc,256]  (aliases xc_out)
    const float* __restrict__ cnt,      // [Nc]
    float* xc_out,                      // [Nc,256]  (same region as ssum)
    float* __restrict__ score)          // [Nc,8]
{
    // LDS tile, row stride 260 (260 % 64 == 4 -> 16 distinct banks for the 16 M-lanes)
    __shared__ float tile[16 * 260];
    const int row0 = blockIdx.x * 16;
    const int t = threadIdx.x;

    // Stage A = ssum / max(cnt,1) into LDS: thread t loads column t of each row.
    #pragma unroll
    for (int r = 0; r < 16; ++r) {
        float inv = 1.0f / fmaxf(cnt[row0 + r], 1.0f);
        tile[r * 260 + t] = ssum[(long)(row0 + r) * HIDDEN + t] * inv;
    }
    __syncthreads();

    const int wave = t >> 5;
    const int lane = t & 31;
    const int m     = lane & 15;          // M (or N for B-frag)
    const int khalf = (lane >> 4) * 2;    // lanes 0-15 -> K 0,1 ; lanes 16-31 -> K 2,3
    const int n0 = wave * 32;             // this wave covers columns [n0, n0+32)

    v8f acc0 = {};
    v8f acc1 = {};

    #pragma unroll 4
    for (int k = 0; k < HIDDEN; k += 4) {
        v2f a;
        a.x = tile[m * 260 + k + khalf];
        a.y = tile[m * 260 + k + khalf + 1];
        const float* Wp = W_lin + (long)(k + khalf) * HIDDEN;
        v2f bA, bB;
        bA.x = Wp[n0 + m];
        bA.y = Wp[HIDDEN + n0 + m];
        bB.x = Wp[n0 + 16 + m];
        bB.y = Wp[HIDDEN + n0 + 16 + m];
        acc0 = __builtin_amdgcn_wmma_f32_16x16x4_f32(
            false, a, false, bA, (short)0, acc0, false, false);
        acc1 = __builtin_amdgcn_wmma_f32_16x16x4_f32(
            false, a, false, bB, (short)0, acc1, false, false);
    }
    __syncthreads();   // all waves done reading the A tile; safe to overwrite LDS

    // Epilogue: bias + relu + residual; write global x_clique_new and LDS copy.
    const int mhi = (lane >> 4) * 8;      // C/D layout: vgpr r -> M = r (+8 for hi lanes)
    #pragma unroll
    for (int ts = 0; ts < 2; ++ts) {
        int n = n0 + ts * 16 + (lane & 15);
        float bias = b_lin[n];
        #pragma unroll
        for (int r = 0; r < 8; ++r) {
            int mm = r + mhi;
            float v = (ts ? acc1[r] : acc0[r]) + bias;
            v = fmaxf(v, 0.0f);
            v += xc_in[(long)(row0 + mm) * HIDDEN + n];
            tile[mm * 260 + n] = v;
            xc_out[(long)(row0 + mm) * HIDDEN + n] = v;
        }
    }
    __syncthreads();

    // Per-head MLP score from LDS: thread t<128 handles (clique c, head h).
    if (t < 128) {
        int c = t >> 3, h = t & 7;
        float xc[HEAD_DIM];
        #pragma unroll
        for (int i = 0; i < HEAD_DIM; ++i) xc[i] = tile[c * 260 + h * HEAD_DIM + i];
        const float* W1h = W1g + h * HEAD_DIM * 64;
        const float* b1h = b1g + h * 64;
        const float* W2h = W2g + h * 64;
        float s = b2g[h];
        for (int k = 0; k < 64; ++k) {
            float hk = b1h[k];
            #pragma unroll
            for (int i = 0; i < HEAD_DIM; ++i) hk = fmaf(xc[i], W1h[i * 64 + k], hk);
            s = fmaf(fmaxf(hk, 0.0f), W2h[k], s);
        }
        score[(long)(row0 + c) * HEADS + h] = s;
    }
}

// ---------------- K3: segment bounds via binary search (clique_batch sorted) ----
__global__ void k3_bounds(const int* __restrict__ batch, int* __restrict__ gstart) {
    int g = blockIdx.x * blockDim.x + threadIdx.x;
    if (g > NUM_GRAPHS) return;
    if (g == NUM_GRAPHS) { gstart[g] = N_CLIQUES; return; }
    int lo = 0, hi = N_CLIQUES;
    while (lo < hi) {
        int mid = (lo + hi) >> 1;
        if (batch[mid] < g) lo = mid + 1; else hi = mid;
    }
    gstart[g] = lo;
}

// ---------------- K4: per-graph softmax + attention-weighted pool ----------------
// One block (256 threads) per graph; no global atomics. Thread t owns output
// channel t (head = t/32). Alpha computed chunk-wise into LDS + written out.
__global__ __launch_bounds__(256) void k4_pool(
    const float* __restrict__ xc,      // [Nc,256] new x_clique
    const float* __restrict__ score,   // [Nc,8]
    const int* __restrict__ gstart,    // [8193]
    float* __restrict__ drug,          // [8192,256]
    float* __restrict__ alpha)         // [Nc,8]
{
    __shared__ float mh[HEADS];
    __shared__ float dh[HEADS];      // 1/(sum exp + 1e-16)
    __shared__ float al[32 * HEADS];
    const int g = blockIdx.x;
    const int s = gstart[g], e = gstart[g + 1];
    const int t = threadIdx.x;

    if (t < HEADS) {
        float m = -3.4028235e38f;
        for (int c = s; c < e; ++c) m = fmaxf(m, score[(long)c * HEADS + t]);
        mh[t] = m;
    }
    __syncthreads();
    if (t < HEADS) {
        float m = mh[t], d = 0.0f;
        for (int c = s; c < e; ++c) d += __expf(score[(long)c * HEADS + t] - m);
        dh[t] = 1.0f / (d + 1e-16f);
    }
    __syncthreads();

    float acc = 0.0f;
    const int h = t >> 5;   // head owning channel t
    for (int c0 = s; c0 < e; c0 += 32) {
        int nch = e - c0; if (nch > 32) nch = 32;
        int cc = t >> 3, hh = t & 7;
        if (cc < nch) {
            float a = __expf(score[(long)(c0 + cc) * HEADS + hh] - mh[hh]) * dh[hh];
            al[cc * HEADS + hh] = a;
            alpha[(long)(c0 + cc) * HEADS + hh] = a;
        }
        __syncthreads();
        for (int c = 0; c < nch; ++c)
            acc = fmaf(xc[(long)(c0 + c) * HIDDEN + t], al[c * HEADS + h], acc);
        __syncthreads();
    }
    drug[(long)g * HIDDEN + t] = acc;
}

// ---------------- launch ----------------
extern "C" void kernel_launch(void* const* d_in, const int* in_sizes, int n_in,
                              void* d_out, int out_size, void* d_ws, size_t ws_size,
                              hipStream_t stream) {
    const float* x        = (const float*)d_in[0];
    const float* x_clique = (const float*)d_in[1];
    const int*   a2c      = (const int*)d_in[2];
    const int*   cbatch   = (const int*)d_in[3];
    // d_in[4] = clique_edge_index: unused by the reference
    const float* W_lin = (const float*)d_in[5];
    const float* b_lin = (const float*)d_in[6];
    const float* W1    = (const float*)d_in[7];
    const float* b1    = (const float*)d_in[8];
    const float* W2    = (const float*)d_in[9];
    const float* b2    = (const float*)d_in[10];

    float* out       = (float*)d_out;
    float* out_drug  = out;                                      // [8192,256]
    float* out_xc    = out + (long)NUM_GRAPHS * HIDDEN;          // [150000,256] (also ssum scratch)
    float* out_alpha = out_xc + (long)N_CLIQUES * HIDDEN;        // [150000,8]

    float* ws     = (float*)d_ws;
    float* cnt    = ws;                         // 150000 floats
    float* score  = ws + 150016;                // 1,200,000 floats
    int*   gstart = (int*)(ws + 150016 + 1200000);  // 8193 ints

    // zero the scatter accumulator (reuses x_clique output region) and counts
    k0_zero<<<2048, 256, 0, stream>>>(out_xc, (long)N_CLIQUES * HIDDEN);
    k0_zero<<<256, 256, 0, stream>>>(cnt, N_CLIQUES);

    k1_scatter<<<N_A2C, 256, 0, stream>>>(x, a2c, out_xc, cnt);

    k2_gemm_score<<<N_CLIQUES / 16, 256, 0, stream>>>(
        x_clique, W_lin, b_lin, W1, b1, W2, b2, out_xc, cnt, out_xc, score);

    k3_bounds<<<(NUM_GRAPHS + 1 + 255) / 256, 256, 0, stream>>>(cbatch, gstart);

    k4_pool<<<NUM_GRAPHS, 256, 0, stream>>>(out_xc, score, gstart, out_drug, out_alpha);
}